// WeightedMSELoss_86371792323054
// MI455X (gfx1250) — compile-verified
//
#include <hip/hip_runtime.h>

// ---------------------------------------------------------------------------
// Weighted MSE loss, MI455X (gfx1250, wave32).
// Pass 1: bandwidth-bound streaming partial sums (b128 NT loads, LDS weight
//         gather, wave32 shuffle reduce) -> d_ws[block].
// Pass 2: single wave reduces block partials; final 32-lane collapse done with
//         V_WMMA_F32_16X16X4_F32 (A = lane accumulators, B = ones).
// Deterministic: fixed grid, fixed reduction order, no atomics.
// ---------------------------------------------------------------------------

typedef float f32x4 __attribute__((ext_vector_type(4)));
typedef int   i32x4 __attribute__((ext_vector_type(4)));
typedef float v2f   __attribute__((ext_vector_type(2)));
typedef float v8f   __attribute__((ext_vector_type(8)));

#define WMSE_BLOCKS  2048
#define WMSE_THREADS 256
#define WAVES_PER_BLOCK (WMSE_THREADS / 32)

__global__ void wmse_partial_kernel(const float* __restrict__ pred,
                                    const int*   __restrict__ lab,
                                    const float* __restrict__ wts,
                                    float*       __restrict__ partials,
                                    int n) {
    __shared__ float w_lds[16];
    __shared__ float wave_sums[WAVES_PER_BLOCK];

    if (threadIdx.x < 10) w_lds[threadIdx.x] = wts[threadIdx.x];
    __syncthreads();

    const int n4     = n >> 2;
    const int gid    = blockIdx.x * blockDim.x + threadIdx.x;
    const int stride = gridDim.x * blockDim.x;

    const f32x4* __restrict__ p4 = (const f32x4*)pred;
    const i32x4* __restrict__ l4 = (const i32x4*)lab;

    float acc = 0.0f;
    for (int i = gid; i < n4; i += stride) {
        // Streaming, non-temporal 128-bit loads: data is touched exactly once.
        f32x4 p = __builtin_nontemporal_load(&p4[i]);
        i32x4 l = __builtin_nontemporal_load(&l4[i]);
#pragma unroll
        for (int c = 0; c < 4; ++c) {
            float d = p[c] - (float)l[c];
            float w = w_lds[l[c]];          // ds_load_b32 gather (10 entries)
            acc = fmaf(w * d, d, acc);
        }
    }
    // Scalar tail (N is a multiple of 4 in practice; kept for safety).
    for (int i = (n4 << 2) + gid; i < n; i += stride) {
        float d = pred[i] - (float)lab[i];
        acc = fmaf(w_lds[lab[i]] * d, d, acc);
    }

    // wave32 tree reduce.
#pragma unroll
    for (int off = 16; off > 0; off >>= 1)
        acc += __shfl_xor(acc, off, 32);

    const int wave = threadIdx.x >> 5;
    const int lane = threadIdx.x & 31;
    if (lane == 0) wave_sums[wave] = acc;
    __syncthreads();

    if (threadIdx.x == 0) {
        float s = 0.0f;
#pragma unroll
        for (int w = 0; w < WAVES_PER_BLOCK; ++w) s += wave_sums[w];
        partials[blockIdx.x] = s;
    }
}

// One wave (32 threads). EXEC is all-ones (no divergence before the WMMA).
__global__ void wmse_final_kernel(const float* __restrict__ partials,
                                  float*       __restrict__ out,
                                  int num_partials, float inv_n) {
    const int lane = threadIdx.x;   // 0..31

    float acc = 0.0f;
    for (int i = lane; i < num_partials; i += 32)
        acc += partials[i];

    // Cross-lane sum via V_WMMA_F32_16X16X4_F32.
    // A (16x4 f32): VGPR0 = {K=0 | lanes 0-15, K=2 | lanes 16-31},
    //               VGPR1 = {K=1, K=3}. So a = {acc, 0} places every lane's
    //               accumulator in A exactly once: A[m][0]=acc[m], A[m][2]=acc[m+16].
    // B (4x16) = ones  =>  D[m][n] = acc[m] + acc[m+16] for all n.
    v2f a; a.x = acc;  a.y = 0.0f;
    v2f b; b.x = 1.0f; b.y = 1.0f;
    v8f c = {};
    v8f d = __builtin_amdgcn_wmma_f32_16x16x4_f32(
        /*neg_a=*/false, a, /*neg_b=*/false, b,
        /*c_mod=*/(short)0, c, /*reuse_a=*/false, /*reuse_b=*/false);

    // D layout: VGPR r, lanes 0-15 -> D[r][lane]; lanes 16-31 -> D[r+8][lane-16].
    // Lane n sums rows 0..7 of its column, lane n+16 sums rows 8..15.
    float s = d[0] + d[1] + d[2] + d[3] + d[4] + d[5] + d[6] + d[7];
    s += __shfl_xor(s, 16, 32);     // combine the two half-column sums

    if (lane == 0) out[0] = s * inv_n;
}

extern "C" void kernel_launch(void* const* d_in, const int* in_sizes, int n_in,
                              void* d_out, int out_size, void* d_ws, size_t ws_size,
                              hipStream_t stream) {
    const float* pred = (const float*)d_in[0];
    const int*   lab  = (const int*)d_in[1];
    const float* wts  = (const float*)d_in[2];
    float*       out  = (float*)d_out;
    float*       part = (float*)d_ws;       // WMSE_BLOCKS floats of scratch
    const int n = in_sizes[0];

    wmse_partial_kernel<<<WMSE_BLOCKS, WMSE_THREADS, 0, stream>>>(pred, lab, wts, part, n);
    wmse_final_kernel<<<1, 32, 0, stream>>>(part, out, WMSE_BLOCKS, 1.0f / (float)n);
}